// DECIFRA_37787122271031
// MI455X (gfx1250) — compile-verified
//
#include <hip/hip_runtime.h>
#include <math.h>

typedef __attribute__((ext_vector_type(2))) float v2f;
typedef __attribute__((ext_vector_type(8))) float v8f;

#define Bn 64
#define Tn 256
#define Cn 64
#define Hn 64
#define HBn 128

// Output layout (floats): logits | matrices | time_logits | predicted | x[:,1:]
#define OFF_MAT  128LL
#define OFF_TL   67108992LL   // 128 + 64*256*4096
#define OFF_PRED 67141760LL   // + 64*256*2
#define OFF_XOUT 68186240LL   // + 64*255*64

__device__ __forceinline__ v8f wmma4(v2f a, v2f b, v8f c) {
  return __builtin_amdgcn_wmma_f32_16x16x4_f32(false, a, false, b, (short)0, c,
                                               false, false);
}

// ---------------------------------------------------------------------------
// D[64xN] = A[64xK] (LDS, lda) * op(B) (+bias)(+relu).
// 8 waves: wave>>1 selects the 16-row M tile, (wave&1) selects which half of
// the N tiles; each wave owns NPW consecutive 16-col tiles so one A fragment
// feeds NPW independent WMMAs per k-step. Depth-1 software pipeline on the
// fragment loads keeps NPW+1 loads in flight across the WMMA burst.
// BT=true : B is [N x K] row-major (torch weight, out = A @ B^T)
// BT=false: B is [K x N] row-major
// ---------------------------------------------------------------------------
template <int NPW, int K, bool BT, bool RELU, bool BIAS>
__device__ __forceinline__ void wgemm_row(const float* __restrict__ A, int lda,
                                          const float* __restrict__ B, int ldb,
                                          const float* __restrict__ bias,
                                          float* __restrict__ D, int ldd,
                                          int wave, int lane) {
  const int lm = lane & 15, lh = lane >> 4;
  const int mt = wave >> 1;
  const int ntb = (wave & 1) * NPW;

  v8f acc[NPW];
#pragma unroll
  for (int i = 0; i < NPW; ++i)
    acc[i] = (v8f){0.f, 0.f, 0.f, 0.f, 0.f, 0.f, 0.f, 0.f};

  const float* Ap = A + (size_t)(mt * 16 + lm) * lda + lh * 2;
  const float* Bt[NPW];
#pragma unroll
  for (int ti = 0; ti < NPW; ++ti) {
    int ncol = (ntb + ti) * 16 + lm;
    Bt[ti] = BT ? (B + (size_t)ncol * ldb + lh * 2)
                : (B + (size_t)(lh * 2) * ldb + ncol);
  }

  v2f a_c, b_c[NPW];
  a_c.x = Ap[0];
  a_c.y = Ap[1];
#pragma unroll
  for (int ti = 0; ti < NPW; ++ti) {
    if (BT) {
      b_c[ti].x = Bt[ti][0];
      b_c[ti].y = Bt[ti][1];
    } else {
      b_c[ti].x = Bt[ti][0];
      b_c[ti].y = Bt[ti][ldb];
    }
  }

#pragma unroll 2
  for (int k = 4; k < K; k += 4) {
    v2f a_n, b_n[NPW];
    a_n.x = Ap[k];
    a_n.y = Ap[k + 1];
#pragma unroll
    for (int ti = 0; ti < NPW; ++ti) {
      if (BT) {
        b_n[ti].x = Bt[ti][k];
        b_n[ti].y = Bt[ti][k + 1];
      } else {
        b_n[ti].x = Bt[ti][(size_t)k * ldb];
        b_n[ti].y = Bt[ti][(size_t)k * ldb + ldb];
      }
    }
#pragma unroll
    for (int ti = 0; ti < NPW; ++ti) acc[ti] = wmma4(a_c, b_c[ti], acc[ti]);
    a_c = a_n;
#pragma unroll
    for (int ti = 0; ti < NPW; ++ti) b_c[ti] = b_n[ti];
  }
#pragma unroll
  for (int ti = 0; ti < NPW; ++ti) acc[ti] = wmma4(a_c, b_c[ti], acc[ti]);

#pragma unroll
  for (int ti = 0; ti < NPW; ++ti) {
    int ncol = (ntb + ti) * 16 + lm;
    float bn = BIAS ? bias[ncol] : 0.0f;
    float* Dp = D + ncol;
#pragma unroll
    for (int v = 0; v < 8; ++v) {
      float val = acc[ti][v] + bn;
      if (RELU) val = fmaxf(val, 0.0f);
      Dp[(size_t)(mt * 16 + v + lh * 8) * ldd] = val;
    }
  }
}

// ---------------------------------------------------------------------------
// Prep: A[j] = embed_w . w_ih[j,:], Bc[j] = embed_b . w_ih[j,:] + b_ih[j];
// also zero the logits accumulator region of d_out.
// ---------------------------------------------------------------------------
__global__ void athena_prep(const float* __restrict__ ew,
                            const float* __restrict__ eb,
                            const float* __restrict__ wih,
                            const float* __restrict__ bih,
                            float* __restrict__ ws, float* __restrict__ logits) {
  int j = threadIdx.x;
  if (j < 192) {
    float a = 0.f, c = 0.f;
    for (int e = 0; e < 32; ++e) {
      a += ew[e] * wih[j * 32 + e];
      c += eb[e] * wih[j * 32 + e];
    }
    ws[j] = a;
    ws[192 + j] = c + bih[j];
  }
  if (j < 128) logits[j] = 0.f;
}

// ---------------------------------------------------------------------------
// Copy x[:,1:,:] to output tail.
// ---------------------------------------------------------------------------
__global__ void athena_xcopy(const float* __restrict__ x,
                             float* __restrict__ xo) {
  int i = blockIdx.x * 256 + threadIdx.x;
  if (i >= Bn * (Tn - 1) * Cn) return;
  int c = i & 63;
  int rest = i >> 6;  // b*(T-1)+t
  int t = rest % (Tn - 1);
  int b = rest / (Tn - 1);
  xo[i] = x[((size_t)b * Tn + t + 1) * Cn + c];
}

// ---------------------------------------------------------------------------
// Main recurrent kernel: one block per batch, t-loop inside.
// ---------------------------------------------------------------------------
__global__ void athena_main(
    const float* __restrict__ x, const float* __restrict__ w_hh,
    const float* __restrict__ b_hh, const float* __restrict__ q1w,
    const float* __restrict__ q1b, const float* __restrict__ q2w,
    const float* __restrict__ q2b, const float* __restrict__ q3w,
    const float* __restrict__ q3b, const float* __restrict__ k1w,
    const float* __restrict__ k1b, const float* __restrict__ k2w,
    const float* __restrict__ k2b, const float* __restrict__ k3w,
    const float* __restrict__ k3b, const float* __restrict__ gbias,
    const float* __restrict__ predw, const float* __restrict__ predbp,
    const float* __restrict__ ws, float* __restrict__ out) {
  extern __shared__ float sm[];
  float* h0 = sm;                  // 64 x 68
  float* h1 = h0 + 64 * 68;        // 64 x 68
  float* t1 = h1 + 64 * 68;        // 64 x 132
  float* t2 = t1 + 64 * 132;       // 64 x 132
  float* t3 = t2 + 64 * 132;       // 64 x 132
  float* tr = t3 + 64 * 132;       // 64 x 68
  float* xs = tr + 64 * 68;        // 64
  float* Ac = xs + 64;             // 192
  float* Bc = Ac + 192;            // 192
  float* pw = Bc + 192;            // 64
  float* red = pw + 64;            // 2
  float* gh = t1;                  // overlay: 64 x 196 spans t1 + part of t2

  const int b = blockIdx.x;
  const int tid = threadIdx.x;
  const int wave = tid >> 5, lane = tid & 31;

  for (int i = tid; i < 64 * 68 * 2; i += 256) sm[i] = 0.f;  // h0 = h1 = 0
  for (int i = tid; i < 384; i += 256) Ac[i] = ws[i];        // Ac then Bc
  if (tid < 64) pw[tid] = predw[tid];
  __syncthreads();

  float* hc = h0;
  float* hn = h1;
  float* matbase = out + OFF_MAT + (size_t)b * Tn * (Cn * Cn);
  float* predbase = out + OFF_PRED + (size_t)b * (Tn - 1) * Cn;
  const float pb = predbp[0];

  // zoff is always 0 but opaque to the compiler: indexing every weight/bias
  // with it makes those loads loop-variant, so LLVM cannot LICM-hoist the
  // fragment loads out of the 256-step t-loop (which previously spilled ~100
  // b64 fragments to scratch). Unlike opaquing the pointers themselves, a
  // scalar offset preserves the global address space -> saddr-form
  // global_load_b64 instead of flat_load_b64 (keeps DScnt decoupled and
  // halves address VGPRs).
  int zoff = 0;

  for (int t = 0; t < Tn; ++t) {
    asm volatile("" : "+s"(zoff));

    if (tid < 64) xs[tid] = x[((size_t)b * Tn + t) * Cn + tid];
    __syncthreads();

    // gh = hc @ w_hh^T + b_hh   (N=192 -> 6 tiles/wave)
    wgemm_row<6, 64, true, false, true>(hc, 68, w_hh + zoff, 64, b_hh + zoff,
                                        gh, 196, wave, lane);
    __syncthreads();

    // GRU gates (gi collapsed to rank-1: x_c*Ac[j] + Bc[j])
    for (int i = tid; i < 4096; i += 256) {
      int c = i >> 6, j = i & 63;
      float xc = xs[c];
      float gr = xc * Ac[j] + Bc[j] + gh[c * 196 + j];
      float gz = xc * Ac[64 + j] + Bc[64 + j] + gh[c * 196 + 64 + j];
      float gn = xc * Ac[128 + j] + Bc[128 + j];
      float ghn = gh[c * 196 + 128 + j];
      float r = 1.f / (1.f + expf(-gr));
      float z = 1.f / (1.f + expf(-gz));
      float n = tanhf(gn + r * ghn);
      float hp = hc[c * 68 + j];
      hc[c * 68 + j] = (1.f - z) * n + z * hp;
    }
    __syncthreads();

    // q MLP -> t3
    wgemm_row<4, 64, true, true, true>(hc, 68, q1w + zoff, 64, q1b + zoff, t1,
                                       132, wave, lane);
    __syncthreads();
    wgemm_row<4, 128, true, true, true>(t1, 132, q2w + zoff, 128, q2b + zoff,
                                        t2, 132, wave, lane);
    __syncthreads();
    wgemm_row<4, 128, true, false, true>(t2, 132, q3w + zoff, 128, q3b + zoff,
                                         t3, 132, wave, lane);
    __syncthreads();
    // k MLP -> t1
    wgemm_row<4, 64, true, true, true>(hc, 68, k1w + zoff, 64, k1b + zoff, t1,
                                       132, wave, lane);
    __syncthreads();
    wgemm_row<4, 128, true, true, true>(t1, 132, k2w + zoff, 128, k2b + zoff,
                                        t2, 132, wave, lane);
    __syncthreads();
    wgemm_row<4, 128, true, false, true>(t2, 132, k3w + zoff, 128, k3b + zoff,
                                         t1, 132, wave, lane);
    __syncthreads();

    // tr = q @ k^T   (N=64 -> 2 tiles/wave)
    wgemm_row<2, 128, true, false, false>(t3, 132, t1, 132, nullptr, tr, 68,
                                          wave, lane);
    __syncthreads();

    // Frobenius norm of tr
    if (tid == 0) red[0] = 0.f;
    __syncthreads();
    float s = 0.f;
    for (int i = tid; i < 4096; i += 256) {
      int c = i >> 6, j = i & 63;
      float v = tr[c * 68 + j];
      s += v * v;
    }
    atomicAdd(&red[0], s);
    __syncthreads();
    float inv = 1.f / sqrtf(red[0]);

    // normalize + gate + write matrices
    float* mo = matbase + (size_t)t * (Cn * Cn);
    for (int i = tid; i < 4096; i += 256) {
      int c = i >> 6, j = i & 63;
      float tv = tr[c * 68 + j] * inv;
      float g = 1.f / (1.f + expf(-(fabsf(tv) + gbias[i])));
      float val = tv * g;
      tr[c * 68 + j] = val;
      mo[i] = val;
    }
    __syncthreads();

    // hn = tr @ hc
    wgemm_row<2, 64, false, false, false>(tr, 68, hc, 68, nullptr, hn, 68,
                                          wave, lane);
    __syncthreads();

    // predictor on post-BTP hidden state
    if (t < Tn - 1 && tid < 64) {
      float sd = 0.f;
      for (int j = 0; j < 64; ++j) sd += hn[tid * 68 + j] * pw[j];
      predbase[(size_t)t * Cn + tid] = sd + pb;
    }
    __syncthreads();

    float* tmp = hc;
    hc = hn;
    hn = tmp;
  }
}

// ---------------------------------------------------------------------------
// Classifier: 16 rows (b,t) per block through 4096->2048->1024->2 MLP.
// ---------------------------------------------------------------------------
__global__ void athena_clf(const float* __restrict__ mat,
                           const float* __restrict__ w1,
                           const float* __restrict__ b1,
                           const float* __restrict__ w2,
                           const float* __restrict__ b2,
                           const float* __restrict__ w3,
                           const float* __restrict__ b3,
                           float* __restrict__ tl,
                           float* __restrict__ logits) {
  extern __shared__ float sm[];
  float* h1 = sm;                 // 16 x 2052
  float* h2 = h1 + 16 * 2052;     // 16 x 1028
  float* chunk = h2;              // overlay: 16 x 516 (used only in layer 1)
  float* red = h2 + 16 * 1028;    // 256

  const int tid = threadIdx.x, wave = tid >> 5, lane = tid & 31;
  const int lm = lane & 15, lh = lane >> 4;
  const int row0 = blockIdx.x * 16;

  // ----- Layer 1: [16,4096] @ W1^T[4096,2048], 16 N-tiles per wave -----
  v8f acc[16];
#pragma unroll
  for (int i = 0; i < 16; ++i)
    acc[i] = (v8f){0.f, 0.f, 0.f, 0.f, 0.f, 0.f, 0.f, 0.f};

  for (int kc = 0; kc < 4096; kc += 512) {
    __syncthreads();
    for (int i = tid; i < 16 * 512; i += 256) {
      int rr = i >> 9, kk = i & 511;
      chunk[rr * 516 + kk] = mat[(size_t)(row0 + rr) * 4096 + kc + kk];
    }
    __syncthreads();
    const float* Ap = chunk + lm * 516 + lh * 2;
    const float* Bp = w1 + (size_t)(wave * 256 + lm) * 4096 + kc + lh * 2;

    v2f a_c, b_c[16];
    a_c.x = Ap[0];
    a_c.y = Ap[1];
#pragma unroll
    for (int ti = 0; ti < 16; ++ti) {
      const float* bp = Bp + (size_t)ti * 16 * 4096;
      b_c[ti].x = bp[0];
      b_c[ti].y = bp[1];
    }
    for (int k = 4; k < 512; k += 4) {
      v2f a_n, b_n[16];
      a_n.x = Ap[k];
      a_n.y = Ap[k + 1];
#pragma unroll
      for (int ti = 0; ti < 16; ++ti) {
        const float* bp = Bp + (size_t)ti * 16 * 4096;
        b_n[ti].x = bp[k];
        b_n[ti].y = bp[k + 1];
      }
#pragma unroll
      for (int ti = 0; ti < 16; ++ti) acc[ti] = wmma4(a_c, b_c[ti], acc[ti]);
      a_c = a_n;
#pragma unroll
      for (int ti = 0; ti < 16; ++ti) b_c[ti] = b_n[ti];
    }
#pragma unroll
    for (int ti = 0; ti < 16; ++ti) acc[ti] = wmma4(a_c, b_c[ti], acc[ti]);
  }
  __syncthreads();
#pragma unroll
  for (int ti = 0; ti < 16; ++ti) {
    int ncol = wave * 256 + ti * 16 + lm;
    float bb = b1[ncol];
#pragma unroll
    for (int v = 0; v < 8; ++v)
      h1[(size_t)(v + lh * 8) * 2052 + ncol] = fmaxf(acc[ti][v] + bb, 0.f);
  }
  __syncthreads();

  // ----- Layer 2: [16,2048] @ W2^T[2048,1024], 8 N-tiles per wave -----
  v8f a2[8];
#pragma unroll
  for (int i = 0; i < 8; ++i)
    a2[i] = (v8f){0.f, 0.f, 0.f, 0.f, 0.f, 0.f, 0.f, 0.f};
  {
    const float* Ap = h1 + lm * 2052 + lh * 2;
    const float* Bp = w2 + (size_t)(wave * 128 + lm) * 2048 + lh * 2;

    v2f a_c, b_c[8];
    a_c.x = Ap[0];
    a_c.y = Ap[1];
#pragma unroll
    for (int ti = 0; ti < 8; ++ti) {
      const float* bp = Bp + (size_t)ti * 16 * 2048;
      b_c[ti].x = bp[0];
      b_c[ti].y = bp[1];
    }
    for (int k = 4; k < 2048; k += 4) {
      v2f a_n, b_n[8];
      a_n.x = Ap[k];
      a_n.y = Ap[k + 1];
#pragma unroll
      for (int ti = 0; ti < 8; ++ti) {
        const float* bp = Bp + (size_t)ti * 16 * 2048;
        b_n[ti].x = bp[k];
        b_n[ti].y = bp[k + 1];
      }
#pragma unroll
      for (int ti = 0; ti < 8; ++ti) a2[ti] = wmma4(a_c, b_c[ti], a2[ti]);
      a_c = a_n;
#pragma unroll
      for (int ti = 0; ti < 8; ++ti) b_c[ti] = b_n[ti];
    }
#pragma unroll
    for (int ti = 0; ti < 8; ++ti) a2[ti] = wmma4(a_c, b_c[ti], a2[ti]);
  }
  __syncthreads();
#pragma unroll
  for (int ti = 0; ti < 8; ++ti) {
    int ncol = wave * 128 + ti * 16 + lm;
    float bb = b2[ncol];
#pragma unroll
    for (int v = 0; v < 8; ++v)
      h2[(size_t)(v + lh * 8) * 1028 + ncol] = fmaxf(a2[ti][v] + bb, 0.f);
  }
  __syncthreads();

  // ----- Layer 3: [16,1024] @ W3^T[1024,2] -----
  {
    int pair = tid >> 3, part = tid & 7;  // 32 (row,out) pairs x 8 partials
    int m = pair >> 1, o = pair & 1;
    float s = 0.f;
    const float* hrow = h2 + (size_t)m * 1028 + part * 128;
    const float* wrow = w3 + (size_t)o * 1024 + part * 128;
    for (int k = 0; k < 128; ++k) s += hrow[k] * wrow[k];
    red[tid] = s;
  }
  __syncthreads();
  if (tid < 32) {
    float s = 0.f;
#pragma unroll
    for (int i = 0; i < 8; ++i) s += red[tid * 8 + i];
    int m = tid >> 1, o = tid & 1;
    float val = s + b3[o];
    int row = row0 + m;
    tl[(size_t)row * 2 + o] = val;
    atomicAdd(&logits[(size_t)(row >> 8) * 2 + o], val * (1.0f / 256.0f));
  }
}

extern "C" void kernel_launch(void* const* d_in, const int* in_sizes, int n_in,
                              void* d_out, int out_size, void* d_ws,
                              size_t ws_size, hipStream_t stream) {
  const float* x = (const float*)d_in[0];
  const float* embed_w = (const float*)d_in[1];
  const float* embed_b = (const float*)d_in[2];
  const float* w_ih = (const float*)d_in[3];
  const float* w_hh = (const float*)d_in[4];
  const float* b_ih = (const float*)d_in[5];
  const float* b_hh = (const float*)d_in[6];
  const float* q1w = (const float*)d_in[7];
  const float* q1b = (const float*)d_in[8];
  const float* q2w = (const float*)d_in[9];
  const float* q2b = (const float*)d_in[10];
  const float* q3w = (const float*)d_in[11];
  const float* q3b = (const float*)d_in[12];
  const float* k1w = (const float*)d_in[13];
  const float* k1b = (const float*)d_in[14];
  const float* k2w = (const float*)d_in[15];
  const float* k2b = (const float*)d_in[16];
  const float* k3w = (const float*)d_in[17];
  const float* k3b = (const float*)d_in[18];
  const float* gbias = (const float*)d_in[19];
  const float* c1w = (const float*)d_in[20];
  const float* c1b = (const float*)d_in[21];
  const float* c2w = (const float*)d_in[22];
  const float* c2b = (const float*)d_in[23];
  const float* c3w = (const float*)d_in[24];
  const float* c3b = (const float*)d_in[25];
  const float* predw = (const float*)d_in[26];
  const float* predb = (const float*)d_in[27];

  float* out = (float*)d_out;
  float* ws = (float*)d_ws;

  athena_prep<<<1, 256, 0, stream>>>(embed_w, embed_b, w_ih, b_ih, ws, out);

  int ncopy = Bn * (Tn - 1) * Cn;
  athena_xcopy<<<(ncopy + 255) / 256, 256, 0, stream>>>(x, out + OFF_XOUT);

  size_t smem_main =
      (size_t)(64 * 68 * 2 + 64 * 132 * 3 + 64 * 68 + 64 + 192 + 192 + 64 + 2) *
      sizeof(float);
  athena_main<<<Bn, 256, smem_main, stream>>>(
      x, w_hh, b_hh, q1w, q1b, q2w, q2b, q3w, q3b, k1w, k1b, k2w, k2b, k3w,
      k3b, gbias, predw, predb, ws, out);

  size_t smem_clf = (size_t)(16 * 2052 + 16 * 1028 + 256) * sizeof(float);
  athena_clf<<<(Bn * Tn) / 16, 256, smem_clf, stream>>>(
      out + OFF_MAT, c1w, c1b, c2w, c2b, c3w, c3b, out + OFF_TL, out);
}